// OnnxNonMaxSuppression_10350871183656
// MI455X (gfx1250) — compile-verified
//
#include <hip/hip_runtime.h>
#include <stdint.h>

#define B_    8
#define N_    16384
#define C_    80
#define MAXK  50
#define TPB   1024
#define SLOTS (N_ / TPB)   // 16 slots per thread

// --------- wave32 u64 max reduction via shuffles (gfx1250 is wave32) ---------
__device__ __forceinline__ unsigned long long wave_max_u64(unsigned long long k) {
#pragma unroll
  for (int m = 16; m >= 1; m >>= 1) {
    unsigned hi = __shfl_xor((unsigned)(k >> 32), m, 32);
    unsigned lo = __shfl_xor((unsigned)(k & 0xffffffffu), m, 32);
    unsigned long long o = ((unsigned long long)hi << 32) | lo;
    k = (o > k) ? o : k;
  }
  return k;
}

__global__ void __launch_bounds__(TPB)
nms_kernel(const float* __restrict__ boxes,      // [B, N, 4]
           const float* __restrict__ scores,     // [B, C, N]
           const int*   __restrict__ mobpc,      // [1]
           const float* __restrict__ iou_thr_p,  // [1]
           const float* __restrict__ score_thr_p,// [1]
           float*       __restrict__ out)        // [B*C*MAXK, 3]
{
  // 64 KB static LDS: score staging buffer, later reused as reduction scratch.
  __shared__ __align__(16) unsigned char s_mem[N_ * 4];
  float*              s_scores = (float*)s_mem;
  unsigned long long* s_red    = (unsigned long long*)s_mem;        // [0..255]
  float4*             s_winbox = (float4*)(s_mem + 256);            // [256..271]
  unsigned long long* s_winkey = (unsigned long long*)(s_mem + 272);// [272..279]

  const int wg   = blockIdx.x;         // 0 .. B_*C_-1
  const int b    = wg / C_;
  const int c    = wg % C_;
  const int t    = threadIdx.x;
  const int wave = t >> 5;
  const int lane = t & 31;

  const float iou_thr   = iou_thr_p[0];
  const float score_thr = score_thr_p[0];
  int kmax = mobpc[0];
  if (kmax > MAXK) kmax = MAXK;
  if (kmax < 0)    kmax = 0;

  // ---- async global -> LDS staging of this workgroup's 64KB score block --------
  // Each lane copies 16B per round, 4 rounds x 1024 threads = 64KB.
  const size_t sbase = ((size_t)b * C_ + c) * (size_t)N_;  // element offset
  {
    const uint32_t lds0 = (uint32_t)(uintptr_t)(&s_mem[0]); // LDS byte offset
    const unsigned char* g0 = (const unsigned char*)(scores + sbase);
#pragma unroll
    for (int r = 0; r < 4; ++r) {
      uint32_t off  = (uint32_t)((r * TPB + t) * 16);
      uint32_t ldsa = lds0 + off;
      uint64_t ga   = (uint64_t)(uintptr_t)(g0 + off);
      asm volatile("global_load_async_to_lds_b128 %0, %1, off"
                   :: "v"(ldsa), "v"(ga) : "memory");
    }
#if __has_builtin(__builtin_amdgcn_s_wait_asynccnt)
    __builtin_amdgcn_s_wait_asynccnt(0);
#else
    asm volatile("s_wait_asynccnt 0" ::: "memory");
#endif
  }
  __syncthreads();  // all waves' async copies complete -> LDS fully populated

  // ---- pull per-thread state into registers -------------------------------
  // Thread t owns slots { t, t+1024, ..., t+15*1024 } (coalesced).
  unsigned sbits[SLOTS];   // float bits of live score, 0 == dead
  float4   bx[SLOTS];
  const float4* boxes4 = (const float4*)boxes + (size_t)b * N_;
#pragma unroll
  for (int j = 0; j < SLOTS; ++j) {
    int slot = t + j * TPB;
    float s  = s_scores[slot];
    sbits[j] = (s > score_thr) ? __float_as_uint(s) : 0u;
    bx[j]    = boxes4[slot];
  }
  __syncthreads();  // everyone consumed s_scores; low bytes now reusable

  // ---- greedy NMS rounds --------------------------------------------------
  int k = 0;
  for (; k < kmax; ++k) {
    // 1) register argmax: key = (score_bits << 32) | ~idx  (ties -> min idx)
    unsigned long long key = 0ull;
#pragma unroll
    for (int j = 0; j < SLOTS; ++j) {
      unsigned long long kk = ((unsigned long long)sbits[j] << 32) |
                              (unsigned)(~(t + j * TPB));
      key = (kk > key) ? kk : key;
    }
    // 2) wave max, then cross-wave max through LDS
    key = wave_max_u64(key);
    if (lane == 0) s_red[wave] = key;
    __syncthreads();
    if (t < 32) {
      unsigned long long kk = wave_max_u64(s_red[t]);
      if (t == 0) *s_winkey = kk;
    }
    __syncthreads();

    unsigned long long wkey = *s_winkey;
    if ((unsigned)(wkey >> 32) == 0u) break;     // no live candidate left
    const int widx = (int)(~(unsigned)(wkey & 0xffffffffu));
    const int wt   = widx & (TPB - 1);
    const int wj   = widx >> 10;

    if (t == 0) {
      float* row = out + ((size_t)wg * MAXK + k) * 3;
      row[0] = (float)b; row[1] = (float)c; row[2] = (float)widx;
    }
    if (t == wt) {            // owner publishes winner box, kills self
      *s_winbox = bx[wj];
      sbits[wj] = 0u;
    }
    __syncthreads();

    // 3) suppress: inter > thr * union   (division-free, union >= 1e-9)
    const float4 w = *s_winbox;
    const float areaW = (w.z - w.x) * (w.w - w.y);
#pragma unroll
    for (int j = 0; j < SLOTS; ++j) {
      if (sbits[j]) {
        float4 a  = bx[j];
        float x1  = fmaxf(a.x, w.x), y1 = fmaxf(a.y, w.y);
        float x2  = fminf(a.z, w.z), y2 = fminf(a.w, w.w);
        float inter = fmaxf(x2 - x1, 0.0f) * fmaxf(y2 - y1, 0.0f);
        float areaA = (a.z - a.x) * (a.w - a.y);
        float uni   = fmaxf(areaW + areaA - inter, 1e-9f);
        if (inter > iou_thr * uni) sbits[j] = 0u;
      }
    }
    // Next round's s_red writes are disjoint from s_winbox/s_winkey bytes,
    // and barriers above order every cross-thread LDS dependency.
  }

  // ---- pad remaining rows with -1 (single writer: no global write races) --
  if (t == 0) {
    for (int kk = k; kk < MAXK; ++kk) {
      float* row = out + ((size_t)wg * MAXK + kk) * 3;
      row[0] = -1.0f; row[1] = -1.0f; row[2] = -1.0f;
    }
  }
}

extern "C" void kernel_launch(void* const* d_in, const int* in_sizes, int n_in,
                              void* d_out, int out_size, void* d_ws, size_t ws_size,
                              hipStream_t stream) {
  const float* boxes     = (const float*)d_in[0];
  const float* scores    = (const float*)d_in[1];
  const int*   mobpc     = (const int*)d_in[2];
  const float* iou_thr   = (const float*)d_in[3];
  const float* score_thr = (const float*)d_in[4];
  float*       out       = (float*)d_out;

  nms_kernel<<<B_ * C_, TPB, 0, stream>>>(boxes, scores, mobpc, iou_thr,
                                          score_thr, out);
}